// RandRotate90_3D_1735166788575
// MI455X (gfx1250) — compile-verified
//
#include <hip/hip_runtime.h>

// RandRotate90_3D: per-sample rot90^k (k = koefs[b] in [0,3]) in the (D,H)
// plane of three [16,128,128,128] fp32 tensors. Zero FLOPs -> pure HBM
// streaming problem (768 MiB total => ~34 us floor at 23.3 TB/s).
// W (128 floats, 512 B) is rotation-invariant: one wave32 handles one row
// with one b128 per lane, so reads and writes are both fully coalesced.
// Non-temporal hints: data is single-touch and larger than the 192 MB L2.
// All row/rotation index math is wave-uniform -> forced into SGPRs via
// readfirstlane + branchless selects (s_cselect, no EXEC divergence).

typedef __attribute__((ext_vector_type(4))) float v4f;

#define ROWS_TOTAL     (16u * 128u * 128u)   // b,d,h rows            = 262144
#define ROWS_PER_BLK   8u                    // 256 threads / wave32
#define VEC_PER_TENSOR (16u << 19)           // 16*128*128*32 float4  = 8388608

__global__ __launch_bounds__(256) void rot90_gather_kernel(
    const v4f* __restrict__ s_vol,
    const v4f* __restrict__ s_msk,
    const v4f* __restrict__ s_skl,
    const int* __restrict__ koefs,
    v4f* __restrict__ dst)
{
    const unsigned tid  = threadIdx.x;
    const unsigned lane = tid & 31u;     // w-vector index 0..31 (only vector value)

    // row is identical across all 32 lanes of a wave -> pin it to an SGPR.
    const unsigned row = __builtin_amdgcn_readfirstlane(
        blockIdx.x * ROWS_PER_BLK + (tid >> 5));        // 0..262143

    const unsigned h = row & 127u;
    const unsigned d = (row >> 7) & 127u;
    const unsigned b = row >> 14;

    // Per-sample rotation count: wave-uniform, keep scalar.
    const int k = __builtin_amdgcn_readfirstlane(koefs[b]);

    // Branchless source-row map (all scalar -> s_cmp + s_cselect):
    //   k=0: (d,h)   k=1: (h,127-d)   k=2: (127-d,127-h)   k=3: (127-h,d)
    const unsigned dn = 127u - d;
    const unsigned hn = 127u - h;
    const unsigned sd = (k == 0) ? d : (k == 1) ? h  : (k == 2) ? dn : hn;
    const unsigned sh = (k == 0) ? h : (k == 1) ? dn : (k == 2) ? hn : d;

    const unsigned srow = (b << 14) | (sd << 7) | sh;   // scalar
    const unsigned sidx = (srow << 5) | lane;           // float4 units
    const unsigned didx = (row  << 5) | lane;           // float4 units

    // Three independent NT b128 streaming loads (MLP), then three NT b128 stores.
    const v4f a = __builtin_nontemporal_load(s_vol + sidx);
    const v4f m = __builtin_nontemporal_load(s_msk + sidx);
    const v4f s = __builtin_nontemporal_load(s_skl + sidx);

    __builtin_nontemporal_store(a, dst + didx);
    __builtin_nontemporal_store(m, dst + VEC_PER_TENSOR     + didx);
    __builtin_nontemporal_store(s, dst + VEC_PER_TENSOR * 2 + didx);
}

extern "C" void kernel_launch(void* const* d_in, const int* in_sizes, int n_in,
                              void* d_out, int out_size, void* d_ws, size_t ws_size,
                              hipStream_t stream) {
    (void)in_sizes; (void)n_in; (void)out_size; (void)d_ws; (void)ws_size;

    const v4f* vol   = (const v4f*)d_in[0];
    const v4f* msk   = (const v4f*)d_in[1];
    const v4f* skl   = (const v4f*)d_in[2];
    const int* koefs = (const int*)d_in[3];
    v4f*       out   = (v4f*)d_out;

    const unsigned blocks = ROWS_TOTAL / ROWS_PER_BLK;  // 32768
    rot90_gather_kernel<<<dim3(blocks), dim3(256), 0, stream>>>(
        vol, msk, skl, koefs, out);
}